// Model_2937757630534
// MI455X (gfx1250) — compile-verified
//
#include <hip/hip_runtime.h>

// GCN 4-layer forward for MI455X (gfx1250, wave32).
// Dense GEMMs use V_WMMA_F32_16X16X4_F32 (full fp32 precision; GEMM is ~1% of
// runtime — ~2.4 GFLOP vs ~3 GB of edge-aggregation traffic, so the chip is
// bound by the scatter at ~130us @ 23.3 TB/s, and the <=51MB agg buffer is
// L2-resident on the 192MB L2). GEMM is templated on (K,M) so the K-loop
// fully unrolls into straight-line global_load_b64 + v_wmma with no EXEC
// manipulation (WMMA requires EXEC all-1s; the M=10 tail is handled by
// clamp+mask, not predication). Store epilogue takes an unguarded fast path
// for full row-tiles (always, since N%16==0). Workspace: 2*N*128 f32
// ping-pong + 3*N f32 + 704 f32 (~104 MB for N=100000).

typedef float v2f __attribute__((ext_vector_type(2)));
typedef float v8f __attribute__((ext_vector_type(8)));

// ---------------- degree / normalization ----------------
__global__ void k_init_deg(float* __restrict__ deg, int N) {
    int i = blockIdx.x * blockDim.x + threadIdx.x;
    if (i < N) deg[i] = 1.0f;                       // self-loop
}

__global__ void k_deg_edges(const int* __restrict__ ei, float* __restrict__ deg, int E) {
    int e = blockIdx.x * blockDim.x + threadIdx.x;
    if (e < E) atomicAdd(&deg[ei[E + e]], 1.0f);    // dst row of edge_index
}

__global__ void k_dinv(const float* __restrict__ deg, float* __restrict__ dinv,
                       float* __restrict__ dinv2, int N) {
    int i = blockIdx.x * blockDim.x + threadIdx.x;
    if (i < N) { float r = rsqrtf(deg[i]); dinv[i] = r; dinv2[i] = r * r; }
}

// ---------------- feature concat [x | pos] -> h0 [N,16] ----------------
__global__ void k_concat(const float* __restrict__ x, const float* __restrict__ pos,
                         float* __restrict__ h, int N) {
    int i = blockIdx.x * blockDim.x + threadIdx.x;
    if (i < N * 16) {
        int n = i >> 4, f = i & 15;
        h[i] = (f < 8) ? x[n * 8 + f] : pos[n * 8 + (f - 8)];
    }
}

// ---------------- dense GEMM: Out[N,M] = H[N,K] * W[K,M] via f32 WMMA -------
// One wave computes one 16x16 tile; K stepped by 4 with V_WMMA_F32_16X16X4_F32.
// A 16x4 layout: lane L -> row (L&15), the two A VGPRs hold K = 2*(L>>4)+{0,1}.
// B mirrored over columns. C/D: 8 VGPRs, acc[i] -> row i + 8*(lane>=16),
// col = lane&15. All loads clamped (never predicated) so EXEC stays all-1s;
// the padded-column case (M=10) zeroes B via a 0/1 multiplier that folds away
// for M % 16 == 0.
template <int K, int M>
__global__ void k_gemm_wmma_f32(const float* __restrict__ H, const float* __restrict__ W,
                                float* __restrict__ Out, int N) {
    constexpr int NTM = (M + 15) / 16;
    int wave = (blockIdx.x * blockDim.x + threadIdx.x) >> 5;
    int lane = threadIdx.x & 31;
    int tm = wave / NTM, tn = wave % NTM;
    if (tm * 16 >= N) return;                       // whole wave exits together

    int r     = lane & 15;
    int khalf = (lane >> 4) << 1;                   // 0 or 2
    int row   = tm * 16 + r;
    int rowc  = row < N ? row : N - 1;              // clamp, keep EXEC full
    int col   = tn * 16 + r;
    int colc;
    float bm;
    if ((M % 16) != 0) { colc = col < M ? col : M - 1; bm = col < M ? 1.0f : 0.0f; }
    else               { colc = col;                   bm = 1.0f; }

    const float* Ap = H + (size_t)rowc * K + khalf;       // +=4 per step
    const float* Bp = W + (size_t)khalf * M + colc;       // +=4*M per step

    v8f acc = {};
#pragma unroll
    for (int k = 0; k < K; k += 4) {
        v2f a, b;
        a.x = Ap[0];                                 // contiguous -> load_b64
        a.y = Ap[1];
        b.x = bm * Bp[0];
        b.y = bm * Bp[M];
        Ap += 4;
        Bp += 4 * M;
        acc = __builtin_amdgcn_wmma_f32_16x16x4_f32(
            /*neg_a=*/false, a, /*neg_b=*/false, b,
            /*c_mod=*/(short)0, acc, /*reuse_a=*/false, /*reuse_b=*/false);
    }

    if (col < M) {
        int rb = tm * 16 + ((lane >> 4) << 3);       // +8 rows for upper half-wave
        float* op = Out + (size_t)rb * M + col;
        if (tm * 16 + 16 <= N) {                     // full tile: unguarded store clause
#pragma unroll
            for (int i = 0; i < 8; ++i) op[(size_t)i * M] = acc[i];
        } else {                                     // ragged tail (unused for N%16==0)
#pragma unroll
            for (int i = 0; i < 8; ++i)
                if (rb + i < N) op[(size_t)i * M] = acc[i];
        }
    }
}

// ---------------- agg init: self-loop term + bias ----------------
__global__ void k_selfloop_bias(const float* __restrict__ Hg, const float* __restrict__ dinv2,
                                const float* __restrict__ b, float* __restrict__ agg,
                                int total, int M) {
    int i = blockIdx.x * blockDim.x + threadIdx.x;
    if (i < total) {
        int n = i / M, f = i - n * M;
        agg[i] = dinv2[n] * Hg[i] + b[f];
    }
}

// ---------------- edge scatter: agg[dst] += dinv[s]*dinv[d]*h[src] ----------
// Consecutive lanes cover consecutive 4-float chunks of the SAME edge row, so
// a 128-wide row is one fully coalesced 512B wave transaction (read + atomic).
__global__ void k_scatter(const int* __restrict__ ei, const float* __restrict__ dinv,
                          const float* __restrict__ Hg, float* __restrict__ agg,
                          int E, int M, int chunksPerEdge) {
    int t = blockIdx.x * blockDim.x + threadIdx.x;
    int e = t / chunksPerEdge;
    int c = t - e * chunksPerEdge;
    if (e >= E) return;
    int s = ei[e], d = ei[E + e];
    float coeff = dinv[s] * dinv[d];
    int f0 = c << 2;
    const float* hp = Hg + (size_t)s * M;
    float* ap = agg + (size_t)d * M;
#pragma unroll
    for (int j = 0; j < 4; ++j) {
        int f = f0 + j;
        if (f < M) atomicAdd(&ap[f], coeff * hp[f]);
    }
}

__global__ void k_relu(float* __restrict__ h, int n) {
    int i = blockIdx.x * blockDim.x + threadIdx.x;
    if (i < n) h[i] = fmaxf(h[i], 0.0f);
}

// ---------------- pooled mean per graph ----------------
__global__ void k_zero(float* __restrict__ p, int n) {
    int i = blockIdx.x * blockDim.x + threadIdx.x;
    if (i < n) p[i] = 0.0f;
}

__global__ void k_pool(const float* __restrict__ h, const int* __restrict__ batch,
                       float* __restrict__ sums, float* __restrict__ cnt, int N) {
    int n = blockIdx.x * blockDim.x + threadIdx.x;
    if (n >= N) return;
    int g = batch[n];
#pragma unroll
    for (int f = 0; f < 10; ++f) atomicAdd(&sums[g * 10 + f], h[(size_t)n * 10 + f]);
    atomicAdd(&cnt[g], 1.0f);
}

__global__ void k_final(const float* __restrict__ sums, const float* __restrict__ cnt,
                        float* __restrict__ out) {
    int i = threadIdx.x;                             // 640 threads
    if (i < 640) out[i] = sums[i] / fmaxf(cnt[i / 10], 1.0f);
}

// ---------------- driver ----------------
static inline int cdiv(int a, int b) { return (a + b - 1) / b; }

template <int K, int M>
static void run_layer(const float* Hin, const float* W, const float* b,
                      float* Hgemm, float* Agg, const int* ei,
                      const float* dinv, const float* dinv2,
                      int N, int E, bool relu, hipStream_t stream) {
    constexpr int NTM = (M + 15) / 16;
    const int tiles = cdiv(N, 16) * NTM;
    k_gemm_wmma_f32<K, M><<<cdiv(tiles * 32, 256), 256, 0, stream>>>(Hin, W, Hgemm, N);
    k_selfloop_bias<<<cdiv(N * M, 256), 256, 0, stream>>>(Hgemm, dinv2, b, Agg, N * M, M);
    const int cpe = cdiv(M, 4);
    k_scatter<<<cdiv(E * cpe, 256), 256, 0, stream>>>(ei, dinv, Hgemm, Agg, E, M, cpe);
    if (relu) k_relu<<<cdiv(N * M, 256), 256, 0, stream>>>(Agg, N * M);
}

extern "C" void kernel_launch(void* const* d_in, const int* in_sizes, int n_in,
                              void* d_out, int out_size, void* d_ws, size_t ws_size,
                              hipStream_t stream) {
    const float* x     = (const float*)d_in[0];
    const float* pos   = (const float*)d_in[1];
    const int*   ei    = (const int*)d_in[2];     // [2,E]: src row then dst row
    const int*   batch = (const int*)d_in[3];
    const float* W1 = (const float*)d_in[4],  *b1 = (const float*)d_in[5];
    const float* W2 = (const float*)d_in[6],  *b2 = (const float*)d_in[7];
    const float* W3 = (const float*)d_in[8],  *b3 = (const float*)d_in[9];
    const float* W4 = (const float*)d_in[10], *b4 = (const float*)d_in[11];

    const int N = in_sizes[0] / 8;                // x is [N,8]
    const int E = in_sizes[2] / 2;                // edge_index is [2,E]

    float* P     = (float*)d_ws;                  // ping buffer  [N,128] max
    float* Q     = P + (size_t)N * 128;           // pong buffer  [N,128] max
    float* deg   = Q + (size_t)N * 128;
    float* dinv  = deg + N;
    float* dinv2 = dinv + N;
    float* sums  = dinv2 + N;                     // [64,10]
    float* cnt   = sums + 640;                    // [64] (contiguous with sums)

    // gcn_norm: deg (with self-loop) -> dinv, dinv^2
    k_init_deg <<<cdiv(N, 256), 256, 0, stream>>>(deg, N);
    k_deg_edges<<<cdiv(E, 256), 256, 0, stream>>>(ei, deg, E);
    k_dinv     <<<cdiv(N, 256), 256, 0, stream>>>(deg, dinv, dinv2, N);

    // h0 = concat(x, pos) into P
    k_concat<<<cdiv(N * 16, 256), 256, 0, stream>>>(x, pos, P, N);

    // 4 GCN layers; each layer: GEMM P->Q, then agg into P (GEMM input dead).
    run_layer<16,  32 >(P, W1, b1, Q, P, ei, dinv, dinv2, N, E, true,  stream);
    run_layer<32,  64 >(P, W2, b2, Q, P, ei, dinv, dinv2, N, E, true,  stream);
    run_layer<64,  128>(P, W3, b3, Q, P, ei, dinv, dinv2, N, E, true,  stream);
    run_layer<128, 10 >(P, W4, b4, Q, P, ei, dinv, dinv2, N, E, false, stream);

    // per-graph mean pool -> d_out [64,10]
    k_zero <<<cdiv(704, 256), 256, 0, stream>>>(sums, 704);   // sums(640)+cnt(64)
    k_pool <<<cdiv(N, 256), 256, 0, stream>>>(P, batch, sums, cnt, N);
    k_final<<<1, 640, 0, stream>>>(sums, cnt, (float*)d_out);
}